// GNNModel_71983651881414
// MI455X (gfx1250) — compile-verified
//
#include <hip/hip_runtime.h>

// GCN: h1 = relu(Agg(x@W1)+b1); h2 = relu(Agg(h1@W2)+b2);
//      h3 = relu(h2@Wfc1+bf1);  out = h3@Wfc2+bf2
// Agg(h)[i] = dinv[i]^2 * h[i] + sum_{e: dst(e)=i} (dinv[src]*ew*dinv[i]) * h[src]
// dinv[i] = rsqrt(1 + sum_{e: dst(e)=i} ew[e])   (self-loop weight 1)

typedef __attribute__((ext_vector_type(2))) float v2f;
typedef __attribute__((ext_vector_type(8))) float v8f;

// ------------------------- CSR build (by destination) -------------------------
__global__ void k_zero(int* __restrict__ p, int n) {
  int i = blockIdx.x * blockDim.x + threadIdx.x;
  if (i < n) p[i] = 0;
}

__global__ void k_count(const int* __restrict__ dst, int* __restrict__ cnt, int e) {
  int i = blockIdx.x * blockDim.x + threadIdx.x;
  if (i < e) atomicAdd(&cnt[dst[i]], 1);
}

// single-block exclusive scan: rowptr[0..n], cursor[i]=rowptr[i]
__global__ void k_scan(const int* __restrict__ cnt, int* __restrict__ rowptr,
                       int* __restrict__ cursor, int n) {
  __shared__ int sm[1024];
  __shared__ int carry;
  if (threadIdx.x == 0) carry = 0;
  __syncthreads();
  for (int base = 0; base < n; base += 1024) {
    int i = base + (int)threadIdx.x;
    int v = (i < n) ? cnt[i] : 0;
    sm[threadIdx.x] = v;
    __syncthreads();
    for (int off = 1; off < 1024; off <<= 1) {
      int t = (threadIdx.x >= (unsigned)off) ? sm[threadIdx.x - off] : 0;
      __syncthreads();
      sm[threadIdx.x] += t;
      __syncthreads();
    }
    int excl = carry + sm[threadIdx.x] - v;
    if (i < n) { rowptr[i] = excl; cursor[i] = excl; }
    __syncthreads();
    if (threadIdx.x == 1023) carry += sm[1023];
    __syncthreads();
  }
  if (threadIdx.x == 0) rowptr[n] = carry;
}

__global__ void k_build(const int* __restrict__ dst, int* __restrict__ cursor,
                        int* __restrict__ eperm, int e) {
  int i = blockIdx.x * blockDim.x + threadIdx.x;
  if (i < e) {
    int p = atomicAdd(&cursor[dst[i]], 1);
    eperm[p] = i;
  }
}

// dinv[i] = rsqrt(1 + sum incident ew)  -- no atomics
__global__ void k_deg_csr(const int* __restrict__ rowptr, const int* __restrict__ eperm,
                          const float* __restrict__ ew, float* __restrict__ dinv, int n) {
  int i = blockIdx.x * blockDim.x + threadIdx.x;
  if (i >= n) return;
  float s = 1.0f;
  const int b = rowptr[i], en = rowptr[i + 1];
  for (int p = b; p < en; ++p) s += ew[eperm[p]];
  dinv[i] = rsqrtf(s);
}

// ------------------------- fp32 WMMA GEMM, LDS-staged weights -------------------------
// out[nrows x NCOLS] = A[nrows x 128] @ W[128 x NCOLS] (+bias)(+relu)
// W is staged transposed into LDS once per block: wt[n][k] = W[k][n]
// (row stride 130 dwords -> ds_load_b64 stays 8B-aligned, banks spread).
//
// V_WMMA_F32_16X16X4_F32 layouts (ISA 7.12.2):
//   A (16x4):  lane l holds M=(l&15); VGPR0=K[(l>>4)*2], VGPR1=K[(l>>4)*2+1]
//   B (4x16):  lane l holds N=(l&15); VGPR0=B[(l>>4)*2][N], VGPR1=B[(l>>4)*2+1][N]
//   C/D (16x16): lane l, VGPR r -> row r + (l>>4)*8, col (l&15)
template <int NCOLS>
__global__ void k_gemm_wmma(const float* __restrict__ A, const float* __restrict__ W,
                            const float* __restrict__ bias, float* __restrict__ out,
                            int nstrips, int do_relu) {
  constexpr int CT = NCOLS / 16;
  __shared__ float wt[NCOLS][130];
  for (int idx = threadIdx.x; idx < 128 * NCOLS; idx += blockDim.x) {
    const int k = idx / NCOLS;
    const int nn = idx % NCOLS;
    wt[nn][k] = W[idx];
  }
  __syncthreads();

  const int wave  = threadIdx.x >> 5;
  const int lane  = threadIdx.x & 31;
  const int strip = blockIdx.x * (blockDim.x >> 5) + wave;
  if (strip >= nstrips) return;
  const int lid  = lane & 15;
  const int half = lane >> 4;
  const int row0 = strip * 16;
  const float* __restrict__ arow = A + (size_t)(row0 + lid) * 128;

  v8f acc[CT] = {};
  for (int k = 0; k < 128; k += 4) {
    const int ka = k + half * 2;                 // even
    const v2f a = *(const v2f*)(arow + ka);      // global_load_b64
#pragma unroll
    for (int ct = 0; ct < CT; ++ct) {
      const v2f b = *(const v2f*)&wt[ct * 16 + lid][ka];  // ds_load_b64
      // 8 args: (neg_a, A, neg_b, B, c_mod, C, reuse_a, reuse_b)
      acc[ct] = __builtin_amdgcn_wmma_f32_16x16x4_f32(
          false, a, false, b, (short)0, acc[ct], false, false);
    }
  }
#pragma unroll
  for (int ct = 0; ct < CT; ++ct) {
    const float bv = bias ? bias[ct * 16 + lid] : 0.0f;
#pragma unroll
    for (int r = 0; r < 8; ++r) {
      const int row = row0 + r + half * 8;
      float v = acc[ct][r] + bv;
      if (do_relu) v = fmaxf(v, 0.0f);
      out[(size_t)row * NCOLS + ct * 16 + lid] = v;
    }
  }
}

// ------------------------- gather aggregation (fused self+bias+relu) -------------------------
// one wave per node; 4 channels per lane (float4); accumulate neighbors in registers
__global__ void k_gather_agg(const int* __restrict__ rowptr, const int* __restrict__ eperm,
                             const int* __restrict__ src, const float* __restrict__ ew,
                             const float* __restrict__ dinv, const float* __restrict__ h,
                             const float* __restrict__ bias, float* __restrict__ out, int n) {
  int t = blockIdx.x * blockDim.x + threadIdx.x;
  int i = t >> 5, lane = t & 31;
  if (i >= n) return;
  const float di = dinv[i];
  const float s2 = di * di;
  const float4 hb = *(const float4*)(h + (size_t)i * 128 + lane * 4);
  float4 acc;
  acc.x = s2 * hb.x; acc.y = s2 * hb.y; acc.z = s2 * hb.z; acc.w = s2 * hb.w;
  const int b = rowptr[i], en = rowptr[i + 1];
  for (int p = b; p < en; ++p) {
    const int e  = eperm[p];                       // uniform across wave
    const int sv = src[e];
    const float nrm = di * ew[e] * dinv[sv];
    const float4 hv = *(const float4*)(h + (size_t)sv * 128 + lane * 4);
    acc.x = fmaf(nrm, hv.x, acc.x);
    acc.y = fmaf(nrm, hv.y, acc.y);
    acc.z = fmaf(nrm, hv.z, acc.z);
    acc.w = fmaf(nrm, hv.w, acc.w);
  }
  const float4 bb = *(const float4*)(bias + lane * 4);
  acc.x = fmaxf(acc.x + bb.x, 0.0f);
  acc.y = fmaxf(acc.y + bb.y, 0.0f);
  acc.z = fmaxf(acc.z + bb.z, 0.0f);
  acc.w = fmaxf(acc.w + bb.w, 0.0f);
  *(float4*)(out + (size_t)i * 128 + lane * 4) = acc;
}

// ------------------------- launch -------------------------
extern "C" void kernel_launch(void* const* d_in, const int* in_sizes, int n_in,
                              void* d_out, int out_size, void* d_ws, size_t ws_size,
                              hipStream_t stream) {
  const float* x   = (const float*)d_in[0];
  const int*   ei  = (const int*)d_in[1];
  const float* ew  = (const float*)d_in[2];
  const float* W1  = (const float*)d_in[3];
  const float* b1  = (const float*)d_in[4];
  const float* W2  = (const float*)d_in[5];
  const float* b2  = (const float*)d_in[6];
  const float* Wf1 = (const float*)d_in[7];
  const float* bf1 = (const float*)d_in[8];
  const float* Wf2 = (const float*)d_in[9];
  const float* bf2 = (const float*)d_in[10];

  const int n = in_sizes[0] / 128;
  const int e = in_sizes[2];
  const int* src = ei;       // edge_index[0]
  const int* dst = ei + e;   // edge_index[1]

  // workspace carve-out (256B aligned segments)
  char* w = (char*)d_ws;
  auto carve = [&](size_t bytes) -> void* {
    void* p = (void*)w;
    w += (bytes + 255) & ~(size_t)255;
    return p;
  };
  float* dinv   = (float*)carve((size_t)n * 4);
  int*   cnt    = (int*)carve((size_t)n * 4);
  int*   rowptr = (int*)carve((size_t)(n + 1) * 4);
  int*   cursor = (int*)carve((size_t)n * 4);
  int*   eperm  = (int*)carve((size_t)e * 4);
  float* bufA   = (float*)carve((size_t)n * 128 * 4);
  float* bufB   = (float*)carve((size_t)n * 128 * 4);

  const int T = 256;
  const int nstrips = (n + 15) / 16;
  const int gGemm = (nstrips + 7) / 8;  // 8 waves (strips) per block
  const int gN    = (n + T - 1) / T;
  const int gE    = (e + T - 1) / T;
  const int gN32  = (int)(((long long)n * 32 + T - 1) / T);

  // CSR by destination (rebuilt every call; cnt explicitly zeroed)
  k_zero<<<gN, T, 0, stream>>>(cnt, n);
  k_count<<<gE, T, 0, stream>>>(dst, cnt, e);
  k_scan<<<1, 1024, 0, stream>>>(cnt, rowptr, cursor, n);
  k_build<<<gE, T, 0, stream>>>(dst, cursor, eperm, e);
  k_deg_csr<<<gN, T, 0, stream>>>(rowptr, eperm, ew, dinv, n);

  // conv1: bufA = x@W1 ; bufB = relu(Agg(bufA) + b1)
  k_gemm_wmma<128><<<gGemm, T, 0, stream>>>(x, W1, nullptr, bufA, nstrips, 0);
  k_gather_agg<<<gN32, T, 0, stream>>>(rowptr, eperm, src, ew, dinv, bufA, b1, bufB, n);

  // conv2: bufA = bufB@W2 ; bufB = relu(Agg(bufA) + b2)
  k_gemm_wmma<128><<<gGemm, T, 0, stream>>>(bufB, W2, nullptr, bufA, nstrips, 0);
  k_gather_agg<<<gN32, T, 0, stream>>>(rowptr, eperm, src, ew, dinv, bufA, b2, bufB, n);

  // fc1 (bias+relu fused): bufA = relu(bufB@Wf1 + bf1)
  k_gemm_wmma<128><<<gGemm, T, 0, stream>>>(bufB, Wf1, bf1, bufA, nstrips, 1);

  // fc2: out = bufA@Wf2 + bf2
  k_gemm_wmma<64><<<gGemm, T, 0, stream>>>(bufA, Wf2, bf2, (float*)d_out, nstrips, 0);
}